// Coarse_layer_34273839022283
// MI455X (gfx1250) — compile-verified
//
#include <hip/hip_runtime.h>
#include <hip/hip_bf16.h>

typedef __attribute__((ext_vector_type(2))) float v2f;
typedef __attribute__((ext_vector_type(4))) float v4f;
typedef __attribute__((ext_vector_type(8))) float v8f;

static constexpr int CEN   = 512;                  // N_CENTROIDS
static constexpr int NB    = CEN * (CEN - 1) / 2;  // 130816 triangular buckets
static constexpr int ECp1  = NB + 1;               // 130817 output slots
static constexpr int HDIM  = 128;
static constexpr int WSTR  = 132;                  // LDS column stride for W^T (bank-conflict-free)

// ---------------------------------------------------------------- utilities
__global__ void fill_f32_kernel(float* __restrict__ p, size_t n, float v) {
    size_t i = (size_t)blockIdx.x * blockDim.x + threadIdx.x;
    if (i < n) p[i] = v;
}
__global__ void fill_f32x4_kernel(v4f* __restrict__ p, size_t n4, float v) {
    size_t i = (size_t)blockIdx.x * blockDim.x + threadIdx.x;
    if (i < n4) { v4f q = {v, v, v, v}; p[i] = q; }
}
__global__ void copy_i2f_kernel(const int* __restrict__ s, float* __restrict__ d, int n) {
    int i = blockIdx.x * blockDim.x + threadIdx.x;
    if (i < n) d[i] = (float)s[i];
}
__global__ void copy_f2f_kernel(const float* __restrict__ s, float* __restrict__ d, int n) {
    int i = blockIdx.x * blockDim.x + threadIdx.x;
    if (i < n) d[i] = s[i];
}

__device__ __forceinline__ int tri_index(int vmin, int vmax) {
    // lexicographically monotone <=> monotone in key vmin*CEN+vmax
    return vmin * CEN - (vmin * (vmin + 1)) / 2 + (vmax - vmin - 1);
}

// ------------------------------------------------------- edge key histogram
__global__ void edge_count_kernel(const int* __restrict__ ei, const int* __restrict__ cci,
                                  int E, unsigned* __restrict__ ecnt,
                                  unsigned* __restrict__ bucketKey) {
    int e = blockIdx.x * blockDim.x + threadIdx.x;
    if (e >= E) return;
    int sv = cci[ei[e]];
    int ev = cci[ei[E + e]];
    if (sv == ev) return;                      // intra-voxel edge -> sentinel (masked out)
    int vmin = sv < ev ? sv : ev;
    int vmax = sv < ev ? ev : sv;
    int bk = tri_index(vmin, vmax);
    atomicAdd(&ecnt[bk], 1u);
    bucketKey[bk] = (unsigned)(vmin * CEN + vmax);   // benign duplicate stores
}

__global__ void node_count_kernel(const int* __restrict__ cci, int N,
                                  unsigned* __restrict__ ncnt) {
    int n = blockIdx.x * blockDim.x + threadIdx.x;
    if (n < N) atomicAdd(&ncnt[cci[n]], 1u);
}

// --------------------------------------------- fused MLP (WMMA) + node pool
// fine[b,n,:] = relu( x[b,n,:] @ W[0:128,:] + d[n]*W[128,:] + bias )
// accumulated straight into nodeSums[b][centroid][h] with f32 atomics.
// W is staged TRANSPOSED in LDS (stride 132) so each lane's B fragment
// (two consecutive K values of one column) is a single aligned ds_load_b64.
__global__ void __launch_bounds__(256)
fine_gemm_pool_kernel(const float* __restrict__ x, const float* __restrict__ Wm,
                      const float* __restrict__ bias, const float* __restrict__ dist,
                      const int* __restrict__ cci, float* __restrict__ nodeSums,
                      int N, int numTiles) {
    extern __shared__ float lds[];            // 128*132 W^T + 128 bias (~68 KB)
    float* ldsWt = lds;                       // ldsWt[col*WSTR + k], k = 0..128
    float* ldsB  = lds + HDIM * WSTR;

    int tid = threadIdx.x;
    for (int idx = tid; idx < 129 * HDIM; idx += 256) {
        int k = idx >> 7, col = idx & 127;    // Wm row-major [129][128]
        ldsWt[col * WSTR + k] = Wm[idx];
    }
    if (tid < HDIM) ldsB[tid] = bias[tid];
    __syncthreads();

    int wave   = tid >> 5;
    int lane   = tid & 31;
    int half   = lane >> 4;      // 0: lanes 0-15, 1: lanes 16-31
    int lane16 = lane & 15;

    int tile = blockIdx.x * 8 + wave;         // wave-uniform -> EXEC stays full
    if (tile >= numTiles) return;
    int bb = blockIdx.y;
    int n0 = tile * 16;

    const float* xbase = x + ((size_t)bb * N + n0) * HDIM;

    // hide epilogue-gather latency behind the K loop
    __builtin_prefetch(&cci[n0], 0, 0);
    __builtin_prefetch(&dist[n0], 0, 0);

    v8f acc[8] = {};                          // 16x128 f32 accumulator tile

    // K = 0..127 in 32 steps of 4 via V_WMMA_F32_16X16X4_F32
    for (int kt = 0; kt < 32; ++kt) {
        int kb = kt * 4 + half * 2;
        const float* ap = xbase + (size_t)lane16 * HDIM + kb;   // A: row=lane16, K=kb..kb+1
        v2f a = *(const v2f*)ap;                                // global_load_b64
#pragma unroll
        for (int nt = 0; nt < 8; ++nt) {
            int col = nt * 16 + lane16;                          // B: col, K=kb..kb+1
            v2f bm = *(const v2f*)&ldsWt[col * WSTR + kb];       // ds_load_b64, no bank conflict
            acc[nt] = __builtin_amdgcn_wmma_f32_16x16x4_f32(
                false, a, false, bm, (short)0, acc[nt], false, false);
        }
    }

    // epilogue: + d[n]*W[128,:] + bias, relu, atomic scatter into centroid sums
    int   ccis[8];
    float dd[8];
#pragma unroll
    for (int v = 0; v < 8; ++v) {
        int node = n0 + v + half * 8;
        int nc   = node < N ? node : N - 1;
        ccis[v] = cci[nc];
        dd[v]   = dist[nc];
    }
#pragma unroll
    for (int nt = 0; nt < 8; ++nt) {
        int   col = nt * 16 + lane16;
        float wl  = ldsWt[col * WSTR + 128];
        float bv  = ldsB[col];
#pragma unroll
        for (int v = 0; v < 8; ++v) {
            int node = n0 + v + half * 8;
            if (node < N) {
                float val = acc[nt][v] + dd[v] * wl + bv;
                val = val > 0.0f ? val : 0.0f;
                atomicAdd(&nodeSums[((size_t)bb * CEN + ccis[v]) * HDIM + col], val);
            }
        }
    }
}

// scalar cleanup for N % 16 != 0 (not hit for N = 100000, kept for generality)
__global__ void fine_remainder_kernel(const float* __restrict__ x, const float* __restrict__ Wm,
                                      const float* __restrict__ bias, const float* __restrict__ dist,
                                      const int* __restrict__ cci, float* __restrict__ nodeSums,
                                      int N, int B, int firstNode) {
    int node = firstNode + blockIdx.x;
    if (node >= N) return;
    int h = threadIdx.x;
    for (int bb = 0; bb < B; ++bb) {
        float s = bias[h];
        const float* xr = x + ((size_t)bb * N + node) * HDIM;
        for (int k = 0; k < HDIM; ++k) s += xr[k] * Wm[k * HDIM + h];
        s += dist[node] * Wm[128 * HDIM + h];
        s = s > 0.0f ? s : 0.0f;
        atomicAdd(&nodeSums[((size_t)bb * CEN + cci[node]) * HDIM + h], s);
    }
}

// ------------------------------------------------- occupancy scan (3 phase)
__global__ void scan_part1_kernel(const unsigned* __restrict__ ecnt,
                                  unsigned* __restrict__ blockTotals) {
    __shared__ unsigned s[256];
    int t = threadIdx.x;
    int b0 = blockIdx.x * 512 + 2 * t;
    unsigned v = 0;
    if (b0 < NB && ecnt[b0]) v++;
    if (b0 + 1 < NB && ecnt[b0 + 1]) v++;
    s[t] = v;
    __syncthreads();
    for (int off = 128; off > 0; off >>= 1) {
        if (t < off) s[t] += s[t + off];
        __syncthreads();
    }
    if (t == 0) blockTotals[blockIdx.x] = s[0];
}

__global__ void scan_part2_kernel(const unsigned* __restrict__ blockTotals,
                                  unsigned* __restrict__ blockOffsets, int nblk) {
    __shared__ unsigned s[256];
    int t = threadIdx.x;
    unsigned v = (t < nblk) ? blockTotals[t] : 0u;
    s[t] = v;
    __syncthreads();
    for (int off = 1; off < 256; off <<= 1) {
        unsigned add = (t >= off) ? s[t - off] : 0u;
        __syncthreads();
        s[t] += add;
        __syncthreads();
    }
    blockOffsets[t] = s[t] - v;                 // exclusive
    if (t == 255) blockOffsets[256] = s[255];   // total unique (informational)
}

__global__ void scan_part3_kernel(const unsigned* __restrict__ ecnt,
                                  const unsigned* __restrict__ blockOffsets,
                                  int* __restrict__ bucketSlot, float* __restrict__ rcnt) {
    __shared__ unsigned s[256];
    int t  = threadIdx.x;
    int b0 = blockIdx.x * 512 + 2 * t;
    unsigned c0 = (b0 < NB) ? ecnt[b0] : 0u;
    unsigned c1 = (b0 + 1 < NB) ? ecnt[b0 + 1] : 0u;
    unsigned o0 = c0 ? 1u : 0u, o1 = c1 ? 1u : 0u;
    unsigned v  = o0 + o1;
    s[t] = v;
    __syncthreads();
    for (int off = 1; off < 256; off <<= 1) {
        unsigned add = (t >= off) ? s[t - off] : 0u;
        __syncthreads();
        s[t] += add;
        __syncthreads();
    }
    unsigned slot0 = blockOffsets[blockIdx.x] + (s[t] - v);
    if (o0) { bucketSlot[b0] = (int)slot0;            rcnt[b0] = 1.0f / (float)c0; }
    if (o1) { bucketSlot[b0 + 1] = (int)(slot0 + o0); rcnt[b0 + 1] = 1.0f / (float)c1; }
}

// --------------------------------------- stream edge_attr once, scatter mean
// 4 edges per 256-thread block; 64 lanes per edge; b128 NT loads (one-shot
// 819 MB stream must not evict the L2-resident accumulation region).
__global__ void __launch_bounds__(256)
edge_scatter_kernel(const int* __restrict__ ei, const int* __restrict__ cci,
                    const float* __restrict__ ea, const int* __restrict__ bucketSlot,
                    const float* __restrict__ rcnt, float* __restrict__ outAttr, int E) {
    int e = blockIdx.x * 4 + (threadIdx.x >> 6);
    if (e >= E) return;
    int sv = cci[ei[e]];
    int ev = cci[ei[E + e]];
    if (sv == ev) return;
    int vmin = sv < ev ? sv : ev;
    int vmax = sv < ev ? ev : sv;
    int bk   = tri_index(vmin, vmax);
    int s    = bucketSlot[bk];
    float r  = rcnt[bk];
    int g  = threadIdx.x & 63;                  // 64 lanes cover B*H = 256 floats
    int bb = g >> 5;                            // batch (B == 2)
    int h4 = g & 31;                            // float4 index within 128-float row
    const v4f* src = (const v4f*)(ea + ((size_t)bb * E + e) * HDIM) + h4;
    v4f val = __builtin_nontemporal_load(src);  // TH=NT: don't pollute L2
    float* dst = outAttr + ((size_t)bb * ECp1 + s) * HDIM + h4 * 4;
    atomicAdd(dst + 0, val.x * r);
    atomicAdd(dst + 1, val.y * r);
    atomicAdd(dst + 2, val.z * r);
    atomicAdd(dst + 3, val.w * r);
}

__global__ void write_edges_kernel(const unsigned* __restrict__ ecnt,
                                   const unsigned* __restrict__ bucketKey,
                                   const int* __restrict__ bucketSlot,
                                   float* __restrict__ outEdges) {
    int bk = blockIdx.x * blockDim.x + threadIdx.x;
    if (bk >= NB || ecnt[bk] == 0) return;
    int s = bucketSlot[bk];
    unsigned key = bucketKey[bk];
    outEdges[s]        = (float)(key / CEN);    // uniq // 512
    outEdges[ECp1 + s] = (float)(key % CEN);    // uniq %  512
}

__global__ void node_norm_kernel(const float* __restrict__ nodeSums,
                                 const unsigned* __restrict__ ncnt,
                                 float* __restrict__ out, int total) {
    int i = blockIdx.x * blockDim.x + threadIdx.x;
    if (i >= total) return;
    int c = (i / HDIM) % CEN;
    float cnt = (float)ncnt[c];
    if (cnt < 1.0f) cnt = 1.0f;
    out[i] = nodeSums[i] / cnt;
}

// ------------------------------------------------------------------ driver
extern "C" void kernel_launch(void* const* d_in, const int* in_sizes, int n_in,
                              void* d_out, int out_size, void* d_ws, size_t ws_size,
                              hipStream_t stream) {
    const float* x    = (const float*)d_in[0];
    const int*   ei   = (const int*)d_in[1];
    const float* ea   = (const float*)d_in[2];
    /* d_in[3] = scale (unused) */
    const int*   cci  = (const int*)d_in[4];
    const float* dist = (const float*)d_in[5];
    const float* Wm   = (const float*)d_in[6];
    const float* bias = (const float*)d_in[7];

    const int N = in_sizes[4];
    const int E = in_sizes[1] / 2;
    const int B = in_sizes[0] / (N * HDIM);

    float* out = (float*)d_out;
    size_t off_attr  = (size_t)B * CEN * HDIM;
    size_t off_edges = off_attr + (size_t)B * ECp1 * HDIM;
    size_t off_cci   = off_edges + 2 * (size_t)ECp1;
    size_t off_dist  = off_cci + (size_t)N;

    // workspace layout (4-byte words, ~2.6 MB)
    unsigned* ecnt         = (unsigned*)d_ws;
    unsigned* bucketKey    = ecnt + NB;
    int*      bucketSlot   = (int*)(bucketKey + NB);
    float*    rcnt         = (float*)(bucketSlot + NB);
    float*    nodeSums     = rcnt + NB;
    unsigned* ncnt         = (unsigned*)(nodeSums + (size_t)B * CEN * HDIM);
    unsigned* blockTotals  = ncnt + CEN;
    unsigned* blockOffsets = blockTotals + 256;
    size_t wsWords = 4 * (size_t)NB + (size_t)B * CEN * HDIM + CEN + 256 + 257;

    const int nscan = (NB + 511) / 512;         // 256

    // 1) zero scratch, init outputs (attrs=0, edges=-1)
    fill_f32_kernel<<<(unsigned)((wsWords + 255) / 256), 256, 0, stream>>>((float*)d_ws, wsWords, 0.0f);
    size_t nAttr4 = (size_t)B * ECp1 * HDIM / 4;
    fill_f32x4_kernel<<<(unsigned)((nAttr4 + 255) / 256), 256, 0, stream>>>((v4f*)(out + off_attr), nAttr4, 0.0f);
    size_t nEdg = 2 * (size_t)ECp1;
    fill_f32_kernel<<<(unsigned)((nEdg + 255) / 256), 256, 0, stream>>>(out + off_edges, nEdg, -1.0f);

    // 2) histograms
    edge_count_kernel<<<(E + 255) / 256, 256, 0, stream>>>(ei, cci, E, ecnt, bucketKey);
    node_count_kernel<<<(N + 255) / 256, 256, 0, stream>>>(cci, N, ncnt);

    // 3) fused MLP (WMMA f32) + centroid pooling
    int numTiles = N / 16;
    size_t shbytes = (size_t)(HDIM * WSTR + HDIM) * sizeof(float);
    dim3 ggrid((numTiles + 7) / 8, B);
    fine_gemm_pool_kernel<<<ggrid, 256, shbytes, stream>>>(x, Wm, bias, dist, cci, nodeSums, N, numTiles);
    int rem = N - numTiles * 16;
    if (rem > 0)
        fine_remainder_kernel<<<rem, HDIM, 0, stream>>>(x, Wm, bias, dist, cci, nodeSums,
                                                        N, B, numTiles * 16);

    // 4) occupancy prefix scan -> bucket->slot map (reproduces sorted unique order)
    scan_part1_kernel<<<nscan, 256, 0, stream>>>(ecnt, blockTotals);
    scan_part2_kernel<<<1, 256, 0, stream>>>(blockTotals, blockOffsets, nscan);
    scan_part3_kernel<<<nscan, 256, 0, stream>>>(ecnt, blockOffsets, bucketSlot, rcnt);

    // 5) single streaming pass over edge_attr (bandwidth-dominant): scatter means
    edge_scatter_kernel<<<(E + 3) / 4, 256, 0, stream>>>(ei, cci, ea, bucketSlot, rcnt, out + off_attr, E);
    write_edges_kernel<<<(NB + 255) / 256, 256, 0, stream>>>(ecnt, bucketKey, bucketSlot, out + off_edges);

    // 6) node means + passthrough outputs
    int totalBCH = B * CEN * HDIM;
    node_norm_kernel<<<(totalBCH + 255) / 256, 256, 0, stream>>>(nodeSums, ncnt, out, totalBCH);
    copy_i2f_kernel<<<(N + 255) / 256, 256, 0, stream>>>(cci, out + off_cci, N);
    copy_f2f_kernel<<<(N + 255) / 256, 256, 0, stream>>>(dist, out + off_dist, N);
}